// DecoderRNN_82832739270674
// MI455X (gfx1250) — compile-verified
//
#include <hip/hip_runtime.h>
#include <math.h>

typedef float v2f __attribute__((ext_vector_type(2)));
typedef float v8f __attribute__((ext_vector_type(8)));

#define V_  32000
#define E_  512
#define H_  512
#define B_  32
#define T_  20
#define OUTROW (21 * V_)   // per-batch output stride

// ---------------------------------------------------------------------------
// One LSTM cell step.
//   gates = x @ W_ih^T + h @ W_hh^T + (b_ih + b_hh)   (B,2048), order i,f,g,o
//   c' = sig(f)*c + sig(i)*tanh(g);  h' = sig(o)*tanh(c')
// Grid: 64 blocks = 32 j-blocks (16 cols each) x 2 batch-halves (16 each).
// Block: 256 threads = 8 waves; wave lane = (batch 0..15, j-parity).
// x and h tiles for the 16 batches are staged in LDS (2 x 32 KB).
// Each lane computes ALL FOUR gate dot products for its (b, j), so the
// nonlinear update is purely local. h/c ping-pong across launches => no race.
// ---------------------------------------------------------------------------
__global__ __launch_bounds__(256)
void lstm_step_kernel(const float* __restrict__ feats,   // used when first==1
                      const int*   __restrict__ caps,    // (B,T) int32
                      const float* __restrict__ emb,     // (V,E)
                      const float* __restrict__ Wih,     // (4H,E)
                      const float* __restrict__ Whh,     // (4H,H)
                      const float* __restrict__ bih,
                      const float* __restrict__ bhh,
                      const float* __restrict__ h_in,
                      const float* __restrict__ c_in,
                      float* __restrict__ h_out,
                      float* __restrict__ c_out,
                      int first, int t)
{
    __shared__ float sx[16 * E_];   // 32 KB
    __shared__ float sh[16 * H_];   // 32 KB

    const int tid  = threadIdx.x;
    const int jblk = blockIdx.x & 31;   // 32 blocks of 16 gate-columns
    const int b0   = (blockIdx.x >> 5) * 16;

    // Stage x (and h) rows for this batch half: 16 rows x 512 = 2048 float4
    for (int i = 0; i < 8; ++i) {
        int e  = tid + i * 256;         // 0..2047
        int r  = e >> 7;                // batch row 0..15
        int c4 = e & 127;               // float4 column
        const float* xsrc;
        if (first) {
            xsrc = feats + (size_t)(b0 + r) * E_;
        } else {
            int row = caps[(b0 + r) * T_ + t];
            xsrc = emb + (size_t)row * E_;
        }
        reinterpret_cast<float4*>(sx)[r * 128 + c4] =
            reinterpret_cast<const float4*>(xsrc)[c4];
        if (!first) {
            reinterpret_cast<float4*>(sh)[r * 128 + c4] =
                reinterpret_cast<const float4*>(h_in + (size_t)(b0 + r) * H_)[c4];
        }
    }
    __syncthreads();

    const int lane = tid & 31;
    const int w    = tid >> 5;
    const int bl   = lane & 15;                 // batch within half
    const int jh   = lane >> 4;                 // j parity within wave
    const int j    = jblk * 16 + w * 2 + jh;    // gate column 0..511

    float ai = bih[j]          + bhh[j];
    float af = bih[H_ + j]     + bhh[H_ + j];
    float ag = bih[2 * H_ + j] + bhh[2 * H_ + j];
    float ao = bih[3 * H_ + j] + bhh[3 * H_ + j];

    // x-part: rows of W_ih are wave-half-uniform, served from L2/WGP$
    {
        const float4* wi = reinterpret_cast<const float4*>(Wih + (size_t)j * E_);
        const float4* wf = reinterpret_cast<const float4*>(Wih + (size_t)(H_ + j) * E_);
        const float4* wg = reinterpret_cast<const float4*>(Wih + (size_t)(2 * H_ + j) * E_);
        const float4* wo = reinterpret_cast<const float4*>(Wih + (size_t)(3 * H_ + j) * E_);
        const float4* xv = reinterpret_cast<const float4*>(sx + bl * E_);
        for (int k = 0; k < E_ / 4; ++k) {
            float4 x = xv[k];
            float4 a = wi[k], b = wf[k], c = wg[k], d = wo[k];
            ai = fmaf(x.x, a.x, fmaf(x.y, a.y, fmaf(x.z, a.z, fmaf(x.w, a.w, ai))));
            af = fmaf(x.x, b.x, fmaf(x.y, b.y, fmaf(x.z, b.z, fmaf(x.w, b.w, af))));
            ag = fmaf(x.x, c.x, fmaf(x.y, c.y, fmaf(x.z, c.z, fmaf(x.w, c.w, ag))));
            ao = fmaf(x.x, d.x, fmaf(x.y, d.y, fmaf(x.z, d.z, fmaf(x.w, d.w, ao))));
        }
    }
    // h-part (skipped on the first step: h == 0)
    if (!first) {
        const float4* wi = reinterpret_cast<const float4*>(Whh + (size_t)j * H_);
        const float4* wf = reinterpret_cast<const float4*>(Whh + (size_t)(H_ + j) * H_);
        const float4* wg = reinterpret_cast<const float4*>(Whh + (size_t)(2 * H_ + j) * H_);
        const float4* wo = reinterpret_cast<const float4*>(Whh + (size_t)(3 * H_ + j) * H_);
        const float4* hv = reinterpret_cast<const float4*>(sh + bl * H_);
        for (int k = 0; k < H_ / 4; ++k) {
            float4 x = hv[k];
            float4 a = wi[k], b = wf[k], c = wg[k], d = wo[k];
            ai = fmaf(x.x, a.x, fmaf(x.y, a.y, fmaf(x.z, a.z, fmaf(x.w, a.w, ai))));
            af = fmaf(x.x, b.x, fmaf(x.y, b.y, fmaf(x.z, b.z, fmaf(x.w, b.w, af))));
            ag = fmaf(x.x, c.x, fmaf(x.y, c.y, fmaf(x.z, c.z, fmaf(x.w, c.w, ag))));
            ao = fmaf(x.x, d.x, fmaf(x.y, d.y, fmaf(x.z, d.z, fmaf(x.w, d.w, ao))));
        }
    }

    const float cprev = first ? 0.0f : c_in[(size_t)(b0 + bl) * H_ + j];
    const float ig = 1.0f / (1.0f + expf(-ai));
    const float fg = 1.0f / (1.0f + expf(-af));
    const float og = 1.0f / (1.0f + expf(-ao));
    const float cn = fg * cprev + ig * tanhf(ag);
    const float hn = og * tanhf(cn);
    c_out[(size_t)(b0 + bl) * H_ + j] = cn;
    h_out[(size_t)(b0 + bl) * H_ + j] = hn;
}

// ---------------------------------------------------------------------------
// Output projection: logits[m=t*32+b][n] = h_seq[m][:] . W_out[n][:] + b_out[n]
// via V_WMMA_F32_16X16X4_F32 (fp32-exact, wave32).
//   Grid: 250 blocks x 128 N-columns. Block: 256 threads = 8 waves,
//   each wave owns a 16-column N-frag and all M (640 rows in 128-row chunks,
//   8 accumulator tiles). A panel staged in LDS (pad 68 -> conflict-free).
// A-frag (16x4, ISA table): lane l: row = l%16, K pair = 2*(l/16)+{0,1}.
// B-frag (4x16): lane l: col = l%16, K pair = 2*(l/16)+{0,1} (symmetric).
// C/D (16x16): VGPR g: M = g + 8*(l/16), N = l%16.
// Also writes the one-hot <start> rows for this block's column slice.
// ---------------------------------------------------------------------------
__global__ __launch_bounds__(256)
void out_proj_kernel(const float* __restrict__ hseq,   // (640,512) row-major
                     const float* __restrict__ Wout,   // (32000,512)
                     const float* __restrict__ bout,   // (32000)
                     float* __restrict__ out)          // (32,21,32000)
{
    constexpr int NB = 128, MC = 128, KC = 64, KPAD = 68;
    __shared__ float sA[MC * KPAD];                    // ~34 KB

    const int tid   = threadIdx.x;
    const int nblk0 = blockIdx.x * NB;

    // <start> one-hot rows: out[b][0][n] = (n==1)
    for (int idx = tid; idx < B_ * NB; idx += 256) {
        int b = idx >> 7;
        int n = nblk0 + (idx & (NB - 1));
        out[(size_t)b * OUTROW + n] = (n == 1) ? 1.0f : 0.0f;
    }

    const int lane = tid & 31;
    const int w    = tid >> 5;
    const int lr   = lane & 15;      // row/col within frag
    const int lh   = lane >> 4;      // lane half -> K pair select
    const int n    = nblk0 + w * 16 + lr;
    const float bias = bout[n];

    for (int mbase = 0; mbase < 640; mbase += MC) {
        v8f acc[MC / 16];
        #pragma unroll
        for (int mt = 0; mt < MC / 16; ++mt) {
            v8f z = {0.f, 0.f, 0.f, 0.f, 0.f, 0.f, 0.f, 0.f};
            acc[mt] = z;
        }

        for (int kbase = 0; kbase < 512; kbase += KC) {
            __syncthreads();
            // stage A panel: 128 rows x 64 cols = 2048 float4
            for (int i = 0; i < 8; ++i) {
                int e  = tid + i * 256;
                int r  = e >> 4;
                int c4 = e & 15;
                float4 v = reinterpret_cast<const float4*>(
                    hseq + (size_t)(mbase + r) * 512 + kbase)[c4];
                *reinterpret_cast<float4*>(&sA[r * KPAD + c4 * 4]) = v;
            }
            __syncthreads();

            const float* wrow = Wout + (size_t)n * 512 + kbase + 2 * lh;
            #pragma unroll 4
            for (int kk = 0; kk < KC; kk += 4) {
                v2f bfrag = *reinterpret_cast<const v2f*>(wrow + kk);
                #pragma unroll
                for (int mt = 0; mt < MC / 16; ++mt) {
                    v2f afrag = *reinterpret_cast<const v2f*>(
                        &sA[(mt * 16 + lr) * KPAD + kk + 2 * lh]);
                    acc[mt] = __builtin_amdgcn_wmma_f32_16x16x4_f32(
                        false, afrag, false, bfrag, (short)0, acc[mt],
                        false, false);
                }
            }
        }

        // scatter accumulators: m = t*32 + b  ->  out[b][t+1][n]
        #pragma unroll
        for (int mt = 0; mt < MC / 16; ++mt) {
            #pragma unroll
            for (int g = 0; g < 8; ++g) {
                int m  = mbase + mt * 16 + g + 8 * lh;
                int tt = m >> 5;
                int b  = m & 31;
                out[(size_t)b * OUTROW + (size_t)(tt + 1) * V_ + n] =
                    acc[mt][g] + bias;
            }
        }
    }
}

// ---------------------------------------------------------------------------
extern "C" void kernel_launch(void* const* d_in, const int* in_sizes, int n_in,
                              void* d_out, int out_size, void* d_ws, size_t ws_size,
                              hipStream_t stream) {
    const float* feats = (const float*)d_in[0];   // (32,512)
    const int*   caps  = (const int*)  d_in[1];   // (32,20)
    const float* emb   = (const float*)d_in[2];   // (32000,512)
    const float* Wih   = (const float*)d_in[3];   // (2048,512)
    const float* Whh   = (const float*)d_in[4];   // (2048,512)
    const float* bih   = (const float*)d_in[5];
    const float* bhh   = (const float*)d_in[6];
    const float* Wout  = (const float*)d_in[7];   // (32000,512)
    const float* bout  = (const float*)d_in[8];
    float* out = (float*)d_out;

    float* hseq = (float*)d_ws;                 // (20,32,512) == GEMM A (640,512)
    float* h0   = hseq + (size_t)T_ * B_ * H_;  // h after features step
    float* c0   = h0 + (size_t)B_ * H_;
    float* c1   = c0 + (size_t)B_ * H_;

    // step from image features (h=c=0)
    lstm_step_kernel<<<64, 256, 0, stream>>>(feats, nullptr, nullptr,
                                             Wih, Whh, bih, bhh,
                                             nullptr, nullptr, h0, c0,
                                             /*first=*/1, /*t=*/0);
    // 20 caption steps; h_seq rows double as the recurrent h buffers
    for (int t = 0; t < T_; ++t) {
        const float* h_in = (t == 0) ? h0 : hseq + (size_t)(t - 1) * B_ * H_;
        const float* c_in = (t & 1) ? c1 : c0;
        float*       c_o  = (t & 1) ? c0 : c1;
        lstm_step_kernel<<<64, 256, 0, stream>>>(nullptr, caps, emb,
                                                 Wih, Whh, bih, bhh,
                                                 h_in, c_in,
                                                 hseq + (size_t)t * B_ * H_, c_o,
                                                 /*first=*/0, t);
    }

    out_proj_kernel<<<V_ / 128, 256, 0, stream>>>(hseq, Wout, bout, out);
}